// QuantumSentenceTransformer_85830626443666
// MI455X (gfx1250) — compile-verified
//
#include <hip/hip_runtime.h>
#include <hip/hip_bf16.h>
#include <math.h>

typedef __attribute__((ext_vector_type(2))) float v2f;
typedef __attribute__((ext_vector_type(8))) float v8f;

#define NQ      4
#define DIMQ    16
#define DFEAT   512
#define WSTRIDE 516  // 516 % 64 = 4 -> bank = 4*n16 + 2*hi + j : perfect 0..63 permutation

// ---------------------------------------------------------------------------
// Kernel A: fold q_params into a 16x16 circuit matrix M, and fold
// Z_SIGNS @ W_post into G (16x2). One wave; lane b propagates basis vector b.
// ---------------------------------------------------------------------------
__global__ void qst_prep(const float* __restrict__ qp,
                         const float* __restrict__ Wpost,
                         float* __restrict__ Mg,
                         float* __restrict__ Gg) {
  const int lane = threadIdx.x;
  if (lane >= DIMQ) return;

  float s[DIMQ];
#pragma unroll
  for (int i = 0; i < DIMQ; ++i) s[i] = (i == lane) ? 1.0f : 0.0f;

  // CNOT perms (wire w acts on bit 3-w):
  // cnot(0,1): p[i] = i ^ (((i>>3)&1) << 2)
  // cnot(2,3): p[i] = i ^  ((i>>1)&1)
  // cnot(1,2): p[i] = i ^ (((i>>2)&1) << 1)
  for (int k = 0; k < 6; ++k) {
    float t[DIMQ];
#pragma unroll
    for (int i = 0; i < DIMQ; ++i) t[i] = s[i ^ (((i >> 3) & 1) << 2)];
#pragma unroll
    for (int i = 0; i < DIMQ; ++i) s[i] = t[i ^ ((i >> 1) & 1)];
#pragma unroll
    for (int i = 0; i < DIMQ; ++i) t[i] = s[i ^ (((i >> 2) & 1) << 1)];
#pragma unroll
    for (int i = 0; i < DIMQ; ++i) s[i] = t[i];

#pragma unroll
    for (int w = 0; w < NQ; ++w) {
      const float h = qp[k * NQ + w] * 0.5f;
      const float c = cosf(h);
      const float sn = sinf(h);
      const int mask = 1 << (3 - w);
#pragma unroll
      for (int i = 0; i < DIMQ; ++i) {
        if ((i & mask) == 0) {
          const int j = i | mask;
          const float a = s[i];
          const float b = s[j];
          s[i] = c * a - sn * b;
          s[j] = sn * a + c * b;
        }
      }
    }
  }

#pragma unroll
  for (int i = 0; i < DIMQ; ++i) Mg[lane * DIMQ + i] = s[i];

  // G = Z_SIGNS @ W_post : G[i][j] = sum_w (1 - 2*bit_{3-w}(i)) * Wpost[w][j]
  float g0 = 0.0f, g1 = 0.0f;
#pragma unroll
  for (int w = 0; w < NQ; ++w) {
    const float z = 1.0f - 2.0f * (float)((lane >> (3 - w)) & 1);
    g0 += z * Wpost[w * 2 + 0];
    g1 += z * Wpost[w * 2 + 1];
  }
  Gg[lane * 2 + 0] = g0;
  Gg[lane * 2 + 1] = g1;
}

// ---------------------------------------------------------------------------
// Kernel B: fused  GEMM(512x4 via WMMA) -> tanh encode -> v@M (WMMA)
//                  -> square -> @G -> out.  8 waves/block, 16 rows/wave.
// ---------------------------------------------------------------------------
__global__ __launch_bounds__(256) void qst_main(
    const float* __restrict__ X, const float* __restrict__ Wpre,
    const float* __restrict__ bpre, const float* __restrict__ bpost,
    const float* __restrict__ Mg, const float* __restrict__ Gg,
    float* __restrict__ out) {
  __shared__ float Wt[DIMQ * WSTRIDE];  // W_pre^T zero-padded to 16 rows, 33 KB
  __shared__ float T[8][DIMQ * DIMQ];   // per-wave transpose tile
  __shared__ float Ms[DIMQ * DIMQ];     // circuit matrix
  __shared__ float Gs[DIMQ * 2];        // folded output matrix

  const int tid = threadIdx.x;
  const int wave = tid >> 5;
  const int lane = tid & 31;
  const int n16 = lane & 15;
  const int hi = lane >> 4;

  // Stage zero-padded W_pre^T (row n = column n of W_pre, n >= 4 -> zeros).
  for (int i = tid; i < DIMQ * WSTRIDE; i += 256) {
    const int n = i / WSTRIDE;
    const int k = i - n * WSTRIDE;
    Wt[i] = (n < NQ && k < DFEAT) ? Wpre[k * NQ + n] : 0.0f;
  }
  if (tid < DIMQ * DIMQ) Ms[tid] = Mg[tid];
  if (tid < DIMQ * 2) Gs[tid] = Gg[tid];
  __syncthreads();

  const long rowBase = ((long)blockIdx.x * 8 + wave) * DIMQ;
  const float* Arow = X + (rowBase + n16) * DFEAT + 2 * hi;
  const int bofs = n16 * WSTRIDE + 2 * hi;  // conflict-free bank permutation

  // ---- GEMM1: pre = X_tile @ W_pre  (K = 512, 128 x wmma_f32_16x16x4_f32)
  v8f acc = {};
#pragma unroll 16
  for (int k0 = 0; k0 < DFEAT; k0 += 4) {
    if ((k0 & 31) == 0)
      __builtin_prefetch(Arow + k0 + 64, 0, 3);  // pull next line toward WGP

    const float2 a2 = *(const float2*)(Arow + k0);
    const float2 b2 = *(const float2*)(&Wt[bofs + k0]);
    v2f a;
    a[0] = a2.x;
    a[1] = a2.y;
    v2f b;
    b[0] = b2.x;
    b[1] = b2.y;
    acc = __builtin_amdgcn_wmma_f32_16x16x4_f32(false, a, false, b, (short)0,
                                                acc, false, false);
  }

  // C layout: lane holds (M = r + 8*hi, N = n16). Transpose through LDS.
#pragma unroll
  for (int r = 0; r < 8; ++r) T[wave][(r + 8 * hi) * DIMQ + n16] = acc[r];
  __syncthreads();

  // ---- Encode: angles -> product state v (both lane halves compute row n16)
  float cp[NQ], cm[NQ];
#pragma unroll
  for (int w = 0; w < NQ; ++w) {
    const float pre = T[wave][n16 * DIMQ + w] + bpre[w];
    const float q = tanhf(pre) * 1.57079632679489662f;
    float sn, c;
    sincosf(q * 0.5f, &sn, &c);
    cp[w] = c + sn;
    cm[w] = c - sn;
  }
  float v[DIMQ];
#pragma unroll
  for (int i = 0; i < DIMQ; ++i) {
    float t = 0.25f;
    t *= ((i >> 3) & 1) ? cp[0] : cm[0];
    t *= ((i >> 2) & 1) ? cp[1] : cm[1];
    t *= ((i >> 1) & 1) ? cp[2] : cm[2];
    t *= (i & 1) ? cp[3] : cm[3];
    v[i] = t;
  }

  // ---- GEMM2: s = v @ M   (16x16x16 via 4 x wmma_f32_16x16x4_f32)
  v8f acc2 = {};
#pragma unroll
  for (int t = 0; t < 4; ++t) {
    const int k0 = 4 * t + 2 * hi;
    v2f a;
    a[0] = v[k0];
    a[1] = v[k0 + 1];
    v2f b;
    b[0] = Ms[k0 * DIMQ + n16];
    b[1] = Ms[(k0 + 1) * DIMQ + n16];
    acc2 = __builtin_amdgcn_wmma_f32_16x16x4_f32(false, a, false, b, (short)0,
                                                 acc2, false, false);
  }

  // ---- probs = s*s ; transpose through LDS ; out = probs @ G + b_post
  __syncthreads();
#pragma unroll
  for (int r = 0; r < 8; ++r) {
    const float x = acc2[r];
    T[wave][(r + 8 * hi) * DIMQ + n16] = x * x;
  }
  __syncthreads();

  if (hi == 0) {
    float o0 = bpost[0], o1 = bpost[1];
#pragma unroll
    for (int n = 0; n < DIMQ; ++n) {
      const float pv = T[wave][n16 * DIMQ + n];
      o0 += pv * Gs[n * 2 + 0];
      o1 += pv * Gs[n * 2 + 1];
    }
    *(float2*)(out + (rowBase + n16) * 2) = make_float2(o0, o1);
  }
}

// ---------------------------------------------------------------------------
extern "C" void kernel_launch(void* const* d_in, const int* in_sizes, int n_in,
                              void* d_out, int out_size, void* d_ws,
                              size_t ws_size, hipStream_t stream) {
  const float* X = (const float*)d_in[0];      // (B, 512)
  const float* Wpre = (const float*)d_in[1];   // (512, 4)
  const float* bpre = (const float*)d_in[2];   // (4)
  const float* qp = (const float*)d_in[3];     // (24)
  const float* Wpost = (const float*)d_in[4];  // (4, 2)
  const float* bpost = (const float*)d_in[5];  // (2)
  float* out = (float*)d_out;                  // (B, 2)

  float* wsf = (float*)d_ws;
  float* Mg = wsf;        // 16x16
  float* Gg = wsf + 256;  // 16x2

  const int B = in_sizes[0] / DFEAT;  // 131072
  qst_prep<<<1, 32, 0, stream>>>(qp, Wpost, Mg, Gg);
  const int blocks = B / 128;  // 8 waves * 16 rows per block
  qst_main<<<blocks, 256, 0, stream>>>(X, Wpre, bpre, bpost, Mg, Gg, out);
}